// LocalMambaBlock_35278861369861
// MI455X (gfx1250) — compile-verified
//
#include <hip/hip_runtime.h>
#include <hip/hip_bf16.h>

typedef float v2f __attribute__((ext_vector_type(2)));
typedef float v8f __attribute__((ext_vector_type(8)));

#define DIM    1024
#define DSTATE 16
#define SEQ    2048
#define BATCH  2
#define TOKENS (BATCH * SEQ)      // 4096
#define NSEG   16
#define LSEG   (SEQ / NSEG)       // 128
#define CHUNK  16                 // timesteps of B/C staged in LDS (scan)

// GEMM block tiling
#define BM 64
#define BN 64
#define BK 16
#define PADA 20                   // As row stride in floats (bank-conflict-free, 16B aligned)
#define PADW 68                   // Ws row stride in floats (bank-conflict-free, 16B aligned)

// ---------------------------------------------------------------------------
// CDNA5 async global->LDS staging (ASYNCcnt-tracked), 16B per lane.
// ---------------------------------------------------------------------------
__device__ __forceinline__ void async_g2l_b128(unsigned lds_byte_off, const float* gaddr) {
    asm volatile("global_load_async_to_lds_b128 %0, %1, off"
                 :: "v"(lds_byte_off), "v"(gaddr) : "memory");
}
__device__ __forceinline__ void wait_asynccnt0() {
    asm volatile("s_wait_asynccnt 0x0" ::: "memory");
}
__device__ __forceinline__ unsigned lds_off_of(const void* p) {
    // generic LDS pointer: addr[31:0] is the LDS byte offset
    return (unsigned)(unsigned long long)p;
}

// ---------------------------------------------------------------------------
// LayerNorm: one block per token, 256 threads, 4 elems/thread.
// ---------------------------------------------------------------------------
__global__ void ln_kernel(const float* __restrict__ x,
                          const float* __restrict__ w,
                          const float* __restrict__ b,
                          float* __restrict__ xn) {
    const int tok = blockIdx.x;
    const int tid = threadIdx.x;
    const float* xr = x + (size_t)tok * DIM;
    float v[4];
    float s = 0.f, ss = 0.f;
#pragma unroll
    for (int i = 0; i < 4; ++i) {
        v[i] = xr[tid + i * 256];
        s += v[i];
        ss += v[i] * v[i];
    }
    __shared__ float rs[256], rss[256];
    rs[tid] = s; rss[tid] = ss;
    __syncthreads();
    for (int off = 128; off > 0; off >>= 1) {
        if (tid < off) { rs[tid] += rs[tid + off]; rss[tid] += rss[tid + off]; }
        __syncthreads();
    }
    const float mu  = rs[0] * (1.0f / DIM);
    const float var = rss[0] * (1.0f / DIM) - mu * mu;
    const float inv = rsqrtf(var + 1e-5f);
    float* xo = xn + (size_t)tok * DIM;
#pragma unroll
    for (int i = 0; i < 4; ++i) {
        const int d = tid + i * 256;
        xo[d] = (v[i] - mu) * inv * w[d] + b[d];
    }
}

// ---------------------------------------------------------------------------
// LDS-tiled fp32 WMMA GEMM:  C[M,N] = A[M,K] * W[K,N] (+bias) (+resid)
// Block: 256 threads (8 waves), 64x64 output tile, BK=16 K-chunks staged in
// LDS via global_load_async_to_lds_b128. Waves: 4(M) x 2(N); each wave does a
// 16x32 slice (2 accumulators sharing one A fragment per K-step).
// Requires M%64==0, N%64==0, K%16==0.
// ---------------------------------------------------------------------------
__global__ void gemm_wmma_tiled(const float* __restrict__ A, int lda,
                                const float* __restrict__ W, int ldw,
                                const float* __restrict__ bias,
                                const float* __restrict__ resid, int ldr,
                                float* __restrict__ Cout, int ldc,
                                int M, int N, int K) {
    __shared__ float As[BM * PADA];   // [row 0..63][k 0..15], padded stride 20
    __shared__ float Ws[BK * PADW];   // [krow 0..15][n 0..63], padded stride 68

    const int tid  = threadIdx.x;
    const int wave = tid >> 5;
    const int lane = tid & 31;
    const int tilesN = N / BN;
    const int bM = (blockIdx.x / tilesN) * BM;
    const int bN = (blockIdx.x % tilesN) * BN;

    const int wm   = wave >> 1;       // 0..3  (M sub-tile)
    const int wn   = wave & 1;        // 0..1  (N sub-tile pair)
    const int m    = lane & 15;
    const int half = lane >> 4;

    // staging assignment: one b128 per thread per tile per chunk
    const int arow = tid >> 2;            // 0..63
    const int acol = (tid & 3) << 2;      // 0,4,8,12
    const int wrow = tid >> 4;            // 0..15
    const int wcol = (tid & 15) << 2;     // 0..60

    const float* gA = A + (size_t)(bM + arow) * lda + acol;
    const float* gW = W + (size_t)wrow * ldw + bN + wcol;
    const unsigned ldsA = lds_off_of(&As[arow * PADA + acol]);
    const unsigned ldsW = lds_off_of(&Ws[wrow * PADW + wcol]);

    const float* fragA = &As[(wm * 16 + m) * PADA];
    const float* fragW = &Ws[wn * 32 + m];

    v8f acc0 = {}, acc1 = {};
    for (int k0 = 0; k0 < K; k0 += BK) {
        __syncthreads();                      // previous chunk fully consumed
        async_g2l_b128(ldsA, gA + k0);
        async_g2l_b128(ldsW, gW + (size_t)k0 * ldw);
        wait_asynccnt0();
        __syncthreads();                      // all waves' tiles visible
#pragma unroll
        for (int ks = 0; ks < BK; ks += 4) {
            const int ka = ks + 2 * half;
            v2f a, b0, b1;
            a[0]  = fragA[ka + 0];
            a[1]  = fragA[ka + 1];
            b0[0] = fragW[(ka + 0) * PADW];
            b0[1] = fragW[(ka + 1) * PADW];
            b1[0] = fragW[(ka + 0) * PADW + 16];
            b1[1] = fragW[(ka + 1) * PADW + 16];
            acc0 = __builtin_amdgcn_wmma_f32_16x16x4_f32(
                false, a, false, b0, (short)0, acc0, false, false);
            acc1 = __builtin_amdgcn_wmma_f32_16x16x4_f32(
                false, a, false, b1, (short)0, acc1, false, false);
        }
    }

    const int col0 = bN + wn * 32 + m;
    const int col1 = col0 + 16;
    const float bv0 = bias ? bias[col0] : 0.0f;
    const float bv1 = bias ? bias[col1] : 0.0f;
#pragma unroll
    for (int r = 0; r < 8; ++r) {
        const int row = bM + wm * 16 + r + 8 * half;
        float v0 = acc0[r] + bv0;
        float v1 = acc1[r] + bv1;
        if (resid) {
            v0 += resid[(size_t)row * ldr + col0];
            v1 += resid[(size_t)row * ldr + col1];
        }
        Cout[(size_t)row * ldc + col0] = v0;
        Cout[(size_t)row * ldc + col1] = v1;
    }
}

// ---------------------------------------------------------------------------
// Skinny fp32 WMMA GEMM for N=16 projections: one 16x16 tile per wave.
// ---------------------------------------------------------------------------
__global__ void gemm_wmma_skinny(const float* __restrict__ A, int lda,
                                 const float* __restrict__ W, int ldw,
                                 const float* __restrict__ bias,
                                 float* __restrict__ Cout, int ldc,
                                 int M, int N, int K) {
    const int wave = (blockIdx.x * blockDim.x + threadIdx.x) >> 5;
    const int lane = threadIdx.x & 31;
    const int tilesN = N >> 4;
    const int tM = wave / tilesN;
    const int tN = wave % tilesN;
    if (tM * 16 >= M) return;

    const int m    = lane & 15;
    const int half = lane >> 4;

    const float* arow = A + (size_t)(tM * 16 + m) * lda;
    const float* wcol = W + (tN * 16 + m);

    v8f acc = {};
#pragma unroll 4
    for (int k = 0; k < K; k += 4) {
        const int ka = k + 2 * half;
        v2f a, bf;
        a[0]  = arow[ka + 0];
        a[1]  = arow[ka + 1];
        bf[0] = wcol[(size_t)(ka + 0) * ldw];
        bf[1] = wcol[(size_t)(ka + 1) * ldw];
        acc = __builtin_amdgcn_wmma_f32_16x16x4_f32(
            false, a, false, bf, (short)0, acc, false, false);
    }

    const int col = tN * 16 + m;
    const float bv = bias ? bias[col] : 0.0f;
#pragma unroll
    for (int r = 0; r < 8; ++r) {
        const int row = tM * 16 + r + 8 * half;
        Cout[(size_t)row * ldc + col] = acc[r] + bv;
    }
}

// ---------------------------------------------------------------------------
// Scan pass 1: per-segment local state (zero init), h only.
// ---------------------------------------------------------------------------
__global__ void scan_local(const float* __restrict__ xz,   // [TOKENS][2*DIM]
                           const float* __restrict__ Bm,   // [TOKENS][DSTATE]
                           const float* __restrict__ Ap,   // [DIM][DSTATE]
                           float* __restrict__ hseg) {     // [B][NSEG][DIM][DSTATE]
    const int tid = threadIdx.x;
    const int db  = blockIdx.x & 3;
    const int seg = (blockIdx.x >> 2) & (NSEG - 1);
    const int b   = blockIdx.x >> 6;
    const int d   = db * 256 + tid;

    float a[DSTATE], h[DSTATE];
#pragma unroll
    for (int n = 0; n < DSTATE; ++n) {
        a[n] = -__expf(Ap[(size_t)d * DSTATE + n]);
        h[n] = 0.0f;
    }

    __shared__ float Bs[CHUNK * DSTATE];
    const int t0 = seg * LSEG;
    for (int c = 0; c < LSEG / CHUNK; ++c) {
        const int tbase = t0 + c * CHUNK;
        __syncthreads();
        Bs[tid] = Bm[((size_t)(b * SEQ + tbase + (tid >> 4))) * DSTATE + (tid & 15)];
        __syncthreads();
#pragma unroll 4
        for (int tt = 0; tt < CHUNK; ++tt) {
            const float x_t = xz[((size_t)(b * SEQ + tbase + tt)) * (2 * DIM) + d];
#pragma unroll
            for (int n = 0; n < DSTATE; ++n)
                h[n] = fmaf(h[n], a[n], x_t * Bs[tt * DSTATE + n]);
        }
    }
    float* hp = hseg + (((size_t)(b * NSEG + seg) * DIM + d) << 4);
#pragma unroll
    for (int n = 0; n < DSTATE; ++n) hp[n] = h[n];
}

// ---------------------------------------------------------------------------
// Scan pass 2: sequential combine over 16 segments per (b,d).
// h_in[s] = A^LSEG * h_in[s-1] + hseg[s-1];  A^128 = exp(128*A_param) (128 even)
// ---------------------------------------------------------------------------
__global__ void scan_prefix(const float* __restrict__ Ap,
                            const float* __restrict__ hseg,
                            float* __restrict__ hin) {
    const int tid = threadIdx.x;
    const int db  = blockIdx.x & 3;
    const int b   = blockIdx.x >> 2;
    const int d   = db * 256 + tid;

    float p[DSTATE], hc[DSTATE];
#pragma unroll
    for (int n = 0; n < DSTATE; ++n) {
        p[n]  = __expf((float)LSEG * Ap[(size_t)d * DSTATE + n]);
        hc[n] = 0.0f;
    }
    for (int s = 0; s < NSEG; ++s) {
        const size_t base = ((size_t)(b * NSEG + s) * DIM + d) << 4;
#pragma unroll
        for (int n = 0; n < DSTATE; ++n) hin[base + n] = hc[n];
#pragma unroll
        for (int n = 0; n < DSTATE; ++n) hc[n] = fmaf(hc[n], p[n], hseg[base + n]);
    }
}

// ---------------------------------------------------------------------------
// Scan pass 3: re-scan with correct h_in, produce gated y:
//   y = (sum_n h*C + x*D) * sigmoid(z)
// ---------------------------------------------------------------------------
__global__ void scan_y(const float* __restrict__ xz,
                       const float* __restrict__ Bm,
                       const float* __restrict__ Cm,
                       const float* __restrict__ Ap,
                       const float* __restrict__ Dv,
                       const float* __restrict__ hin,
                       float* __restrict__ yg) {          // [TOKENS][DIM]
    const int tid = threadIdx.x;
    const int db  = blockIdx.x & 3;
    const int seg = (blockIdx.x >> 2) & (NSEG - 1);
    const int b   = blockIdx.x >> 6;
    const int d   = db * 256 + tid;

    float a[DSTATE], h[DSTATE];
    const float* hp = hin + (((size_t)(b * NSEG + seg) * DIM + d) << 4);
#pragma unroll
    for (int n = 0; n < DSTATE; ++n) {
        a[n] = -__expf(Ap[(size_t)d * DSTATE + n]);
        h[n] = hp[n];
    }
    const float Dd = Dv[d];

    __shared__ float Bs[CHUNK * DSTATE];
    __shared__ float Cs[CHUNK * DSTATE];
    const int t0 = seg * LSEG;
    for (int c = 0; c < LSEG / CHUNK; ++c) {
        const int tbase = t0 + c * CHUNK;
        __syncthreads();
        {
            const size_t idx = ((size_t)(b * SEQ + tbase + (tid >> 4))) * DSTATE + (tid & 15);
            Bs[tid] = Bm[idx];
            Cs[tid] = Cm[idx];
        }
        __syncthreads();
#pragma unroll 2
        for (int tt = 0; tt < CHUNK; ++tt) {
            const size_t rowx = ((size_t)(b * SEQ + tbase + tt)) * (2 * DIM);
            const float x_t = xz[rowx + d];
            const float z_t = xz[rowx + DIM + d];
            float y = 0.0f;
#pragma unroll
            for (int n = 0; n < DSTATE; ++n) {
                h[n] = fmaf(h[n], a[n], x_t * Bs[tt * DSTATE + n]);
                y    = fmaf(h[n], Cs[tt * DSTATE + n], y);
            }
            y = (y + x_t * Dd) / (1.0f + __expf(-z_t));   // * sigmoid(z)
            yg[((size_t)(b * SEQ + tbase + tt)) * DIM + d] = y;
        }
    }
}

// ---------------------------------------------------------------------------
// Launch
// ---------------------------------------------------------------------------
extern "C" void kernel_launch(void* const* d_in, const int* in_sizes, int n_in,
                              void* d_out, int out_size, void* d_ws, size_t ws_size,
                              hipStream_t stream) {
    const float* x      = (const float*)d_in[0];
    const float* norm_w = (const float*)d_in[1];
    const float* norm_b = (const float*)d_in[2];
    const float* W_in   = (const float*)d_in[3];
    const float* b_in   = (const float*)d_in[4];
    const float* W_B    = (const float*)d_in[5];
    const float* b_B    = (const float*)d_in[6];
    const float* W_C    = (const float*)d_in[7];
    const float* b_C    = (const float*)d_in[8];
    const float* A_p    = (const float*)d_in[9];
    const float* Dv     = (const float*)d_in[10];
    const float* W_out  = (const float*)d_in[11];
    const float* b_out  = (const float*)d_in[12];
    float* out = (float*)d_out;

    // Workspace layout
    char* ws = (char*)d_ws;
    float* xn   = (float*)(ws);                                   // 4096*1024  (reused as yg)
    float* xz   = (float*)(ws + (size_t)TOKENS * DIM * 4);        // 4096*2048
    float* Bm   = (float*)(ws + (size_t)TOKENS * DIM * 4 + (size_t)TOKENS * 2 * DIM * 4);
    float* Cm   = (float*)((char*)Bm + (size_t)TOKENS * DSTATE * 4);
    float* hseg = (float*)((char*)Cm + (size_t)TOKENS * DSTATE * 4);
    float* hin  = (float*)((char*)hseg + (size_t)BATCH * NSEG * DIM * DSTATE * 4);
    float* yg   = xn;  // xn dead after GEMM1

    // 1. LayerNorm
    ln_kernel<<<TOKENS, 256, 0, stream>>>(x, norm_w, norm_b, xn);

    // 2. xz = xn @ W_in + b_in   (M=4096, N=2048, K=1024)
    gemm_wmma_tiled<<<(TOKENS / BM) * ((2 * DIM) / BN), 256, 0, stream>>>(
        xn, DIM, W_in, 2 * DIM, b_in, nullptr, 0, xz, 2 * DIM,
        TOKENS, 2 * DIM, DIM);

    // 3/4. B_mat, C_mat = x_ssm @ W_B/C + b  (A = xz with lda=2048, N=16)
    {
        const int waves = (TOKENS / 16) * (DSTATE / 16);
        gemm_wmma_skinny<<<(waves + 7) / 8, 256, 0, stream>>>(
            xz, 2 * DIM, W_B, DSTATE, b_B, Bm, DSTATE, TOKENS, DSTATE, DIM);
        gemm_wmma_skinny<<<(waves + 7) / 8, 256, 0, stream>>>(
            xz, 2 * DIM, W_C, DSTATE, b_C, Cm, DSTATE, TOKENS, DSTATE, DIM);
    }

    // 5-7. Segmented scan (local -> prefix-combine -> rescan+gate)
    scan_local <<<BATCH * NSEG * (DIM / 256), 256, 0, stream>>>(xz, Bm, A_p, hseg);
    scan_prefix<<<BATCH * (DIM / 256),        256, 0, stream>>>(A_p, hseg, hin);
    scan_y     <<<BATCH * NSEG * (DIM / 256), 256, 0, stream>>>(xz, Bm, Cm, A_p, Dv, hin, yg);

    // 8. out = x + yg @ W_out + b_out   (M=4096, N=1024, K=1024)
    gemm_wmma_tiled<<<(TOKENS / BM) * (DIM / BN), 256, 0, stream>>>(
        yg, DIM, W_out, DIM, b_out, x, DIM, out, DIM,
        TOKENS, DIM, DIM);
}